// LocalClustering_30477087933108
// MI455X (gfx1250) — compile-verified
//
#include <hip/hip_runtime.h>

// ---------------------------------------------------------------------------
// LocalClustering for MI455X (gfx1250, wave32).
//   prep_keys   : float -> order-preserving u32 keys
//   rank_sort   : stable per-segment rank sort; tiles staged via CDNA5 async
//                 global->LDS (b128/b32, ASYNCcnt), uint4 LDS scan
//   cluster_seg : B independent serial waves (one per segment, LDS bitmap)
//   scan_counts : prefix-scan of per-segment center counts -> rs_new
//   fixup       : relocate sel, pad -1, add segment offsets to ggather
//   (fallback: exact-global-order single-wave cluster_serial if ws too small)
// ---------------------------------------------------------------------------

#define MAX_SEGS   64
#define TILE_WORDS 2048   // 8KB LDS staging tile for sort keys

// ---- CDNA5 async global->LDS copies (ASYNCcnt path), inline asm ISA §10 ----
__device__ __forceinline__ void async_copy_u32(const unsigned* g, unsigned* l) {
#if defined(__gfx1250__)
  unsigned           lds_off = (unsigned)(__UINTPTR_TYPE__)l;  // low 32 = LDS offset
  unsigned long long ga      = (unsigned long long)(__UINTPTR_TYPE__)g;
  asm volatile("global_load_async_to_lds_b32 %0, %1, off"
               :: "v"(lds_off), "v"(ga) : "memory");
#else
  *l = *g;
#endif
}

__device__ __forceinline__ void async_copy_b128(const unsigned* g, unsigned* l) {
#if defined(__gfx1250__)
  unsigned           lds_off = (unsigned)(__UINTPTR_TYPE__)l;
  unsigned long long ga      = (unsigned long long)(__UINTPTR_TYPE__)g;
  asm volatile("global_load_async_to_lds_b128 %0, %1, off"
               :: "v"(lds_off), "v"(ga) : "memory");
#else
  *(uint4*)l = *(const uint4*)g;
#endif
}

__device__ __forceinline__ void async_join() {
#if defined(__gfx1250__)
  asm volatile("s_wait_asynccnt 0" ::: "memory");
#endif
}

// ---- helpers ---------------------------------------------------------------
__device__ __forceinline__ int find_seg(const int* rs, int nseg, int idx) {
  int s = 0;
  while (s + 1 < nseg && idx >= rs[s + 1]) ++s;
  return s;
}

// Kernel: float -> order-preserving unsigned key.
__global__ __launch_bounds__(256)
void prep_keys(const float* __restrict__ hier, unsigned* __restrict__ keys, int V) {
  int i = blockIdx.x * 256 + threadIdx.x;
  if (i >= V) return;
  unsigned u = __float_as_uint(hier[i]);
  unsigned m = (unsigned)((int)u >> 31) | 0x80000000u;
  keys[i] = u ^ m;
}

// Kernel: stable per-segment rank sort (== jnp.lexsort((hier, seg_ids))).
// rank(i) = #{ j in seg(i) : k[j] < k[i] || (k[j]==k[i] && j<i) }
__global__ __launch_bounds__(256)
void rank_sort(const unsigned* __restrict__ keys,
               const int* __restrict__ row_splits,
               int nseg, int V, int* __restrict__ order) {
  __shared__ uint4 tk4[TILE_WORDS / 4];
  __shared__ int rs[MAX_SEGS + 1];
  unsigned* tk = (unsigned*)tk4;
  const int tid = threadIdx.x;
  const int i0  = blockIdx.x * 256;
  const int i   = i0 + tid;

  if (tid <= nseg && tid <= MAX_SEGS) rs[tid] = row_splits[tid];
  __syncthreads();

  const int iq = (i < V) ? i : (V - 1);
  const int sm = find_seg(rs, nseg, iq);
  const int s0 = rs[sm], s1 = rs[sm + 1];
  const unsigned ki = keys[iq];

  const int lastI = min(i0 + 255, V - 1);
  const int lo = rs[find_seg(rs, nseg, i0)];
  const int hi = rs[find_seg(rs, nseg, lastI) + 1];

  const bool v4ok =
      ((((unsigned long long)(__UINTPTR_TYPE__)keys) & 15ull) == 0) && ((lo & 3) == 0);

  int rank = 0;
  for (int tb = lo; tb < hi; tb += TILE_WORDS) {
    const int tlen = min(TILE_WORDS, hi - tb);

    if (v4ok) {                                    // 16B async DMA chunks
      const int nq = tlen >> 2;
      for (int q = tid; q < nq; q += 256)
        async_copy_b128(keys + tb + 4 * q, tk + 4 * q);
      for (int e = (nq << 2) + tid; e < tlen; e += 256)
        async_copy_u32(keys + tb + e, tk + e);     // <=3 tail words
    } else {
      for (int e = tid; e < tlen; e += 256)
        async_copy_u32(keys + tb + e, tk + e);
    }
    async_join();                                   // s_wait_asynccnt 0
    __syncthreads();

    const int jb = max(tb, s0) - tb;
    const int je = min(tb + tlen, s1) - tb;
    int j = jb;
    for (; j < je && (j & 3); ++j) {
      const unsigned kj = tk[j];
      rank += (kj < ki || (kj == ki && (tb + j) < iq)) ? 1 : 0;
    }
    for (; j + 3 < je; j += 4) {                    // ds_load_b128, 4 cmp/access
      const uint4 q = tk4[j >> 2];
      const int jg = tb + j;
      rank += (q.x < ki || (q.x == ki && (jg + 0) < iq)) ? 1 : 0;
      rank += (q.y < ki || (q.y == ki && (jg + 1) < iq)) ? 1 : 0;
      rank += (q.z < ki || (q.z == ki && (jg + 2) < iq)) ? 1 : 0;
      rank += (q.w < ki || (q.w == ki && (jg + 3) < iq)) ? 1 : 0;
    }
    for (; j < je; ++j) {
      const unsigned kj = tk[j];
      rank += (kj < ki || (kj == ki && (tb + j) < iq)) ? 1 : 0;
    }
    __syncthreads();
  }
  if (i < V) order[s0 + rank] = i;
}

// Kernel: per-segment serial greedy absorb (neighbours are in-segment by
// construction, and the global visit order is segment-major, so the B chains
// are independent). One wave32 per segment, private LDS bitmap.
__global__ __launch_bounds__(32)
void cluster_seg(const int* __restrict__ order,
                 const int* __restrict__ neighs,
                 const int* __restrict__ row_splits,
                 int V, int K,
                 int* __restrict__ lsel,      // local centers at seg_start+j
                 int* __restrict__ rs_new,    // [b+1] <- local count
                 int* __restrict__ ggather) { // local indices (offset later)
  extern __shared__ unsigned bitmap[];
  const int lane = threadIdx.x;
  const int b    = blockIdx.x;
  const int seg0 = row_splits[b], seg1 = row_splits[b + 1];
  const unsigned segLen = (unsigned)(seg1 - seg0);
  const int words = (int)((segLen + 31u) >> 5);

  for (int w = lane; w < words; w += 32) bitmap[w] = 0u;
  __syncthreads();

  int c = 0;
  int vnext = (seg0 < seg1) ? order[seg0] : 0;
  for (int i = seg0; i < seg1; ++i) {
    const int v = vnext;
    if (i + 1 < seg1) vnext = order[i + 1];
    if (lane < 2)                                   // warm next 256B neigh row
      __builtin_prefetch(&neighs[(long long)vnext * K + lane * 32], 0, 0);

    int flag = 0;
    if (lane == 0) {
      const unsigned li  = (unsigned)(v - seg0);
      const unsigned m   = 1u << (li & 31);
      const unsigned old = atomicOr(&bitmap[li >> 5], m);   // ds_or_rtn_b32
      flag = (old & m) ? 0 : 1;
    }
    flag = __builtin_amdgcn_readfirstlane(flag);

    if (flag) {                                     // v becomes a center
      if (lane == 0) { lsel[seg0 + c] = v; ggather[v] = c; }
      const long long base = (long long)v * K;
      for (int k = lane; k < K; k += 32) {          // wave32: 2 lockstep passes
        const int n = neighs[base + k];
        int nn = (n >= 0) ? n : v;                  // -1 pad -> self noop
        unsigned li = (unsigned)(nn - seg0);
        if (li >= segLen) { nn = v; li = (unsigned)(v - seg0); }  // safety clamp
        const unsigned m   = 1u << (li & 31);
        const unsigned old = atomicOr(&bitmap[li >> 5], m);
        if (!(old & m)) ggather[nn] = c;            // pre-op value == "free"
      }
      ++c;
    }
  }
  if (lane == 0) rs_new[b + 1] = c;
}

// Kernel: exclusive prefix over per-segment counts -> rs_new.
__global__ void scan_counts(int* __restrict__ rs_new, int nseg) {
  if (blockIdx.x == 0 && threadIdx.x == 0) {
    rs_new[0] = 0;
    int run = 0;
    for (int s = 0; s < nseg; ++s) { run += rs_new[s + 1]; rs_new[s + 1] = run; }
  }
}

// Kernel: sel[rs_new[s]+j] = lsel[rs[s]+j], pad tail -1, ggather += seg offset.
__global__ __launch_bounds__(256)
void fixup(const int* __restrict__ lsel,
           const int* __restrict__ row_splits,
           const int* __restrict__ rs_new_in,
           int nseg, int V,
           int* __restrict__ sel, int* __restrict__ ggather) {
  __shared__ int rs[MAX_SEGS + 1];
  __shared__ int rn[MAX_SEGS + 1];
  const int tid = threadIdx.x;
  if (tid <= nseg && tid <= MAX_SEGS) { rs[tid] = row_splits[tid]; rn[tid] = rs_new_in[tid]; }
  __syncthreads();

  const int i = blockIdx.x * 256 + tid;
  if (i >= V) return;

  const int total = rn[nseg];
  if (i >= total) {
    sel[i] = -1;
  } else {
    int s = 0;
    while (s + 1 < nseg && i >= rn[s + 1]) ++s;
    sel[i] = lsel[rs[s] + (i - rn[s])];
  }
  ggather[i] += rn[find_seg(rs, nseg, i)];          // localize -> global index
}

// ---- Fallback path (exact global visit order, used when ws is too small) ---
__global__ __launch_bounds__(32)
void cluster_serial(const int* __restrict__ order,
                    const int* __restrict__ neighs,
                    const int* __restrict__ row_splits,
                    int V, int K, int nseg,
                    int* __restrict__ sel,
                    int* __restrict__ rs_new,
                    int* __restrict__ ggather) {
  extern __shared__ unsigned bitmap[];
  __shared__ int rs[MAX_SEGS + 1];
  __shared__ int cnt[MAX_SEGS];
  const int lane  = threadIdx.x;
  const int words = (V + 31) >> 5;
  for (int w = lane; w < words; w += 32) bitmap[w] = 0u;
  for (int s = lane; s <= nseg; s += 32) { rs[s] = row_splits[s]; if (s < nseg) cnt[s] = 0; }
  __syncthreads();

  int count = 0;
  for (int i = 0; i < V; ++i) {
    if (lane == 0 && (i & 63) == 0)
      __builtin_prefetch(&order[min(i + 256, V - 1)], 0, 0);
    const int v = order[i];
    int flag = 0;
    if (lane == 0) {
      const unsigned m   = 1u << (v & 31);
      const unsigned old = atomicOr(&bitmap[v >> 5], m);
      flag = (old & m) ? 0 : 1;
    }
    flag = __builtin_amdgcn_readfirstlane(flag);
    if (flag) {
      if (lane == 0) { sel[count] = v; ggather[v] = count; cnt[find_seg(rs, nseg, v)] += 1; }
      const long long base = (long long)v * K;
      for (int k = lane; k < K; k += 32) {
        const int n = neighs[base + k];
        const int nn = (n >= 0) ? n : v;
        const unsigned m   = 1u << (nn & 31);
        const unsigned old = atomicOr(&bitmap[nn >> 5], m);
        if (!(old & m)) ggather[nn] = count;
      }
      ++count;
    }
  }
  __syncthreads();
  if (lane == 0) {
    int run = 0;
    rs_new[0] = 0;
    for (int s = 0; s < nseg; ++s) { run += cnt[s]; rs_new[s + 1] = run; }
  }
}

__global__ __launch_bounds__(256)
void finalize_sel(int* __restrict__ sel, const int* __restrict__ total_ptr, int V) {
  int i = blockIdx.x * 256 + threadIdx.x;
  if (i >= V) return;
  if (i >= *total_ptr) sel[i] = -1;
}

// ---------------------------------------------------------------------------
extern "C" void kernel_launch(void* const* d_in, const int* in_sizes, int n_in,
                              void* d_out, int out_size, void* d_ws, size_t ws_size,
                              hipStream_t stream) {
  const int*   neighs     = (const int*)d_in[0];
  const float* hier       = (const float*)d_in[1];
  const int*   row_splits = (const int*)d_in[2];

  const int V    = in_sizes[1];            // 262144
  const int K    = in_sizes[0] / V;        // 64
  const int nseg = in_sizes[2] - 1;        // 4

  int* out     = (int*)d_out;              // int32 as raw 32-bit
  int* sel     = out;                      // [V]
  int* rs_new  = out + V;                  // [nseg+1]
  int* ggather = out + V + nseg + 1;       // [V]

  const int    blocks   = (V + 255) / 256;
  const size_t bm_bytes = (size_t)((V + 31) / 32) * sizeof(unsigned);

  if (ws_size >= (size_t)V * 12u && nseg >= 1 && nseg <= MAX_SEGS) {
    // Parallel path: keys | order | lsel in scratch.
    unsigned* keys  = (unsigned*)d_ws;
    int*      order = (int*)d_ws + V;
    int*      lsel  = (int*)d_ws + 2 * (size_t)V;

    prep_keys<<<blocks, 256, 0, stream>>>(hier, keys, V);
    rank_sort<<<blocks, 256, 0, stream>>>(keys, row_splits, nseg, V, order);
    cluster_seg<<<nseg, 32, bm_bytes, stream>>>(order, neighs, row_splits,
                                                V, K, lsel, rs_new, ggather);
    scan_counts<<<1, 1, 0, stream>>>(rs_new, nseg);
    fixup<<<blocks, 256, 0, stream>>>(lsel, row_splits, rs_new, nseg, V,
                                      sel, ggather);
  } else {
    // Fallback: alias output regions (proven safe orderings, see round 1).
    unsigned* keys;
    int*      order;
    if (ws_size >= (size_t)V * 8u) {
      keys  = (unsigned*)d_ws;
      order = (int*)d_ws + V;
    } else {
      keys  = (unsigned*)ggather;   // consumed before ggather is written
      order = sel;                  // cluster writes sel[count<=i] behind reads
    }
    prep_keys<<<blocks, 256, 0, stream>>>(hier, keys, V);
    rank_sort<<<blocks, 256, 0, stream>>>(keys, row_splits, nseg, V, order);
    cluster_serial<<<1, 32, bm_bytes, stream>>>(order, neighs, row_splits,
                                                V, K, nseg, sel, rs_new, ggather);
    finalize_sel<<<blocks, 256, 0, stream>>>(sel, rs_new + nseg, V);
  }
}